// SparseLambdaAttentionLayer_72378788873003
// MI455X (gfx1250) — compile-verified
//
#include <hip/hip_runtime.h>
#include <hip/hip_bf16.h>

// Fused SparseLambdaAttention for MI455X (gfx1250, wave32).
// One workgroup (256 threads = 8 waves) per batch. All intermediates in LDS.
// GEMMs via V_WMMA_F32_16X16X4_F32 (fp32-exact), chained over K in steps of 4.

typedef __attribute__((ext_vector_type(2))) float v2f;
typedef __attribute__((ext_vector_type(8))) float v8f;

#define TOPK 10
#define NEG_INF (-3.402823466e38f)

// One 16x16 output tile D[r0..r0+15][c0..c0+15] = A * B, K-dim reduced in
// steps of 4 with f32 WMMA.  A is accessed as A[r*ars + c*acs] (so a
// transposed operand is just ars=1, acs=rowlen).  B is row-major K x N.
__device__ __forceinline__ void mma_tile_f32(
    const float* __restrict__ A, int ars, int acs,
    const float* __restrict__ B, int ldb,
    float* __restrict__ D, int ldd,
    int r0, int c0, int K, int lane)
{
    const int ml = lane & 15;   // M (for A) / N (for B,D) index within tile
    const int kh = lane >> 4;   // K-half selector
    v8f acc = {0.f, 0.f, 0.f, 0.f, 0.f, 0.f, 0.f, 0.f};
    for (int kk = 0; kk < K; kk += 4) {
        const int k0 = kk + 2 * kh;
        v2f a, b;
        a.x = A[(r0 + ml) * ars + (k0 + 0) * acs];
        a.y = A[(r0 + ml) * ars + (k0 + 1) * acs];
        b.x = B[(k0 + 0) * ldb + (c0 + ml)];
        b.y = B[(k0 + 1) * ldb + (c0 + ml)];
        // (neg_a, A, neg_b, B, c_mod, C, reuse_a, reuse_b)
        acc = __builtin_amdgcn_wmma_f32_16x16x4_f32(
            false, a, false, b, (short)0, acc, false, false);
    }
#pragma unroll
    for (int r = 0; r < 8; ++r)
        D[(r0 + r + 8 * kh) * ldd + (c0 + ml)] = acc[r];
}

__global__ __launch_bounds__(256)
void SparseLambdaAttention_gfx1250_kernel(
    const float* __restrict__ gF,   // [B,64,32]
    const float* __restrict__ gC,   // [B,64,32]
    const float* __restrict__ gWq,  // [32,32]
    const float* __restrict__ gWk,  // [32,32]
    const float* __restrict__ gWv,  // [32,64]
    float* __restrict__ gOut)       // [B,640,32]
{
    const int b    = blockIdx.x;
    const int tid  = threadIdx.x;
    const int lane = tid & 31;
    const int wave = tid >> 5;

    __shared__ float sF [64 * 32];   // featureVec slab
    __shared__ float sC [64 * 32];   // contextVec slab
    __shared__ float sWq[32 * 32];
    __shared__ float sWk[32 * 32];
    __shared__ float sWv[32 * 64];
    __shared__ float sQ [64 * 32];   // Q = F*Wq
    __shared__ float sK [64 * 32];   // softmax(C*Wk) over M
    __shared__ float sVW[64 * 64];   // V = C*Wv, later overwritten by weight
    __shared__ float sLam[32 * 64];  // lam = K^T * V
    __shared__ float sTopW[64 * TOPK];
    __shared__ int   sTopI[64 * TOPK];

    // ---- Phase 0: stage inputs (coalesced float4 loads) --------------------
    {
        const float4* f4 = (const float4*)(gF + (size_t)b * 2048);
        const float4* c4 = (const float4*)(gC + (size_t)b * 2048);
        for (int i = tid; i < 512; i += 256) ((float4*)sF)[i] = f4[i];
        for (int i = tid; i < 512; i += 256) ((float4*)sC)[i] = c4[i];
        if (tid < 256) ((float4*)sWq)[tid] = ((const float4*)gWq)[tid];
        if (tid < 256) ((float4*)sWk)[tid] = ((const float4*)gWk)[tid];
        for (int i = tid; i < 512; i += 256) ((float4*)sWv)[i] = ((const float4*)gWv)[i];
    }
    __syncthreads();

    // ---- Phase 1: Q = F*Wq, Kpre = C*Wk, V = C*Wv (WMMA, 32 tiles / 8 waves)
    {
        const int ti = wave >> 1, tj = wave & 1;          // 8 Q tiles, 8 K tiles
        mma_tile_f32(sF, 32, 1, sWq, 32, sQ, 32, ti * 16, tj * 16, 32, lane);
        mma_tile_f32(sC, 32, 1, sWk, 32, sK, 32, ti * 16, tj * 16, 32, lane);
#pragma unroll
        for (int q = 0; q < 2; ++q) {                     // 16 V tiles
            const int vt = wave * 2 + q;
            mma_tile_f32(sC, 32, 1, sWv, 64, sVW, 64,
                         (vt >> 2) * 16, (vt & 3) * 16, 32, lane);
        }
    }
    __syncthreads();

    // ---- Phase 2: softmax over M (64 rows) per k-channel -------------------
    if (tid < 32) {
        const int k = tid;
        float mx = NEG_INF;
        for (int m = 0; m < 64; ++m) mx = fmaxf(mx, sK[m * 32 + k]);
        float s = 0.f;
        for (int m = 0; m < 64; ++m) {
            float e = expf(sK[m * 32 + k] - mx);
            sK[m * 32 + k] = e;
            s += e;
        }
        const float inv = 1.0f / s;
        for (int m = 0; m < 64; ++m) sK[m * 32 + k] *= inv;
    }
    __syncthreads();

    // ---- Phase 3: lam = K^T * V  (32x64, 8 tiles / 8 waves, Kdim = 64) -----
    // A = sK transposed: elem(r,c) = sK[c*32 + r]  ->  ars=1, acs=32
    mma_tile_f32(sK, 1, 32, sVW, 64, sLam, 64,
                 (wave >> 2) * 16, (wave & 3) * 16, 64, lane);
    __syncthreads();

    // ---- Phase 4: weight = Q * lam (64x64, overwrites V buffer) ------------
#pragma unroll
    for (int q = 0; q < 2; ++q) {
        const int wt = wave * 2 + q;
        mma_tile_f32(sQ, 32, 1, sLam, 64, sVW, 64,
                     (wt >> 2) * 16, (wt & 3) * 16, 32, lane);
    }
    __syncthreads();

    // ---- Phase 5: per-row top-10 + softmax ---------------------------------
    if (tid < 64) {
        const int n = tid;
        float* wrow = sVW + n * 64;   // this row is private to thread n
        float cval[TOPK];
        int   cidx[TOPK];
#pragma unroll
        for (int t = 0; t < TOPK; ++t) {
            float best = NEG_INF;
            int   bi   = 0;
            for (int m = 0; m < 64; ++m) {
                const float v = wrow[m];
                if (v > best) { best = v; bi = m; }   // lowest index on ties
            }
            cval[t] = best;
            cidx[t] = bi;
            wrow[bi] = NEG_INF;
        }
        const float mx = cval[0];
        float s = 0.f;
        float e[TOPK];
#pragma unroll
        for (int t = 0; t < TOPK; ++t) { e[t] = expf(cval[t] - mx); s += e[t]; }
        const float inv = 1.0f / s;
#pragma unroll
        for (int t = 0; t < TOPK; ++t) {
            sTopW[n * TOPK + t] = e[t] * inv;
            sTopI[n * TOPK + t] = cidx[t];
        }
    }
    __syncthreads();

    // ---- Phase 6: out[n,t,:] = w * F[n,:] * C[idx,:]  (coalesced float4) ---
    float4* out4 = (float4*)(gOut + (size_t)b * 64 * TOPK * 32);
    for (int s = tid; s < 64 * TOPK * 8; s += 256) {
        const int n  = s / (TOPK * 8);
        const int r  = s - n * (TOPK * 8);
        const int t  = r >> 3;
        const int dc = r & 7;
        const float w = sTopW[n * TOPK + t];
        const int   m = sTopI[n * TOPK + t];
        const float4 f = ((const float4*)(sF + n * 32))[dc];
        const float4 c = ((const float4*)(sC + m * 32))[dc];
        float4 o;
        o.x = w * f.x * c.x;
        o.y = w * f.y * c.y;
        o.z = w * f.z * c.z;
        o.w = w * f.w * c.w;
        out4[s] = o;
    }
}

extern "C" void kernel_launch(void* const* d_in, const int* in_sizes, int n_in,
                              void* d_out, int out_size, void* d_ws, size_t ws_size,
                              hipStream_t stream) {
    const float* gF  = (const float*)d_in[0];  // [B,64,32]
    const float* gC  = (const float*)d_in[1];  // [B,64,32]
    const float* gWq = (const float*)d_in[2];  // [32,32]
    const float* gWk = (const float*)d_in[3];  // [32,32]
    const float* gWv = (const float*)d_in[4];  // [32,64]
    float* gOut = (float*)d_out;               // [B,640,32]

    const int B = in_sizes[0] / (64 * 32);     // 256 for the reference setup
    SparseLambdaAttention_gfx1250_kernel<<<B, 256, 0, stream>>>(
        gF, gC, gWq, gWk, gWv, gOut);
}